// HybridQuantumClassicalModel_72748156059661
// MI455X (gfx1250) — compile-verified
//
#include <hip/hip_runtime.h>
#include <math.h>

typedef __attribute__((ext_vector_type(2))) float v2f;
typedef __attribute__((ext_vector_type(8))) float v8f;

#define NQ 9
#define DIM 512
#define WAVES 8

// d_ws layout (floats):
//  [0,512)        wc[k] = sum_i SIGNS[i,k]*fc_w[i]
//  [512,3584)     4 matrices (mi = layer*2 + side; side0=qubits0-3, side1=qubits5-8)
//                 each: UrT_fmt(256) | UiT_fmt(256) | (-UiT)_fmt(256), B-operand lane layout
//  [3584,3616)    per-layer 2x2 rot gates: layer*16 + {rot3:0, rot4:8}, 8 floats each (complex 2x2)
#define WS_MAT 512
#define WS_ROT 3584

__device__ __forceinline__ v8f wmma4(v2f a, v2f b, v8f c) {
  // D(16x16,f32) = A(16x4,f32) x B(4x16,f32) + C
  return __builtin_amdgcn_wmma_f32_16x16x4_f32(false, a, false, b, (short)0, c, false, false);
}

// rot = RY(ty) * RX(tx) as complex 2x2: {g00re,g00im,g01re,g01im,g10re,g10im,g11re,g11im}
__device__ __forceinline__ void rot_gate(float tx, float ty, float* g) {
  float cx = cosf(0.5f * tx), sx = sinf(0.5f * tx);
  float cy = cosf(0.5f * ty), sy = sinf(0.5f * ty);
  g[0] =  cy * cx; g[1] =  sy * sx;
  g[2] = -sy * cx; g[3] = -cy * sx;
  g[4] =  sy * cx; g[5] = -cy * sx;
  g[6] =  cy * cx; g[7] = -sy * sx;
}

// ---------------- setup: build batch-invariant gate matrices ----------------
__global__ void qc_setup_kernel(const float* __restrict__ w,
                                const float* __restrict__ fc_w,
                                float* __restrict__ ws) {
  int t = threadIdx.x;
  // wc[k] = sum_i (1-2*bit_i(k)) * fc_w[i]   (qubit i <-> bit 8-i)
  for (int k = t; k < DIM; k += blockDim.x) {
    float acc = 0.f;
    for (int i = 0; i < NQ; ++i)
      acc += (((k >> (8 - i)) & 1) ? -fc_w[i] : fc_w[i]);
    ws[k] = acc;
  }
  if (t < 4) {  // 2x2 rotations for qubits 3 and 4, per layer
    int layer = t >> 1, q = 3 + (t & 1);
    float g[8];
    rot_gate(w[(2 * layer) * NQ + q], w[(2 * layer + 1) * NQ + q], g);
    float* o = ws + WS_ROT + layer * 16 + (t & 1) * 8;
    for (int i = 0; i < 8; ++i) o[i] = g[i];
  }
  if (t >= 4 && t < 8) {  // build the four 16x16 complex unitaries
    int mi = t - 4, layer = mi >> 1, side = mi & 1;
    float ur[16][16], ui[16][16];
    for (int r = 0; r < 16; ++r)
      for (int c = 0; c < 16; ++c) { ur[r][c] = (r == c) ? 1.f : 0.f; ui[r][c] = 0.f; }
    // CX ladder within the nibble: control = higher bit (lower qubit index)
    for (int s = 0; s < 3; ++s) {
      int pc = 3 - s, pt = 2 - s;
      for (int r = 0; r < 16; ++r)
        if (((r >> pc) & 1) && !((r >> pt) & 1)) {
          int rr = r ^ (1 << pt);
          for (int c = 0; c < 16; ++c) {
            float a = ur[r][c]; ur[r][c] = ur[rr][c]; ur[rr][c] = a;
            float b = ui[r][c]; ui[r][c] = ui[rr][c]; ui[rr][c] = b;
          }
        }
    }
    // rotations: side0 -> qubits 0,1,2 (bits 3,2,1); side1 -> qubits 5..8 (bits 3..0)
    int nrot = side ? 4 : 3;
    for (int s = 0; s < nrot; ++s) {
      int q = side ? (5 + s) : s;
      int p = 3 - s;
      float g[8];
      rot_gate(w[(2 * layer) * NQ + q], w[(2 * layer + 1) * NQ + q], g);
      for (int c = 0; c < 16; ++c)
        for (int r0 = 0; r0 < 16; ++r0)
          if (!((r0 >> p) & 1)) {
            int r1 = r0 | (1 << p);
            float are = ur[r0][c], aim = ui[r0][c];
            float bre = ur[r1][c], bim = ui[r1][c];
            ur[r0][c] = g[0] * are - g[1] * aim + g[2] * bre - g[3] * bim;
            ui[r0][c] = g[0] * aim + g[1] * are + g[2] * bim + g[3] * bre;
            ur[r1][c] = g[4] * are - g[5] * aim + g[6] * bre - g[7] * bim;
            ui[r1][c] = g[4] * aim + g[5] * are + g[6] * bim + g[7] * bre;
          }
    }
    // Pre-format B = U^T into WMMA B-operand lane layout:
    // chunk j, vgpr v: lanes 0-15 -> B[4j+v][lane], lanes 16-31 -> B[4j+v+2][lane-16]
    float* base = ws + WS_MAT + mi * 768;
    for (int f = 0; f < 256; ++f) {
      int j = f >> 6, v = (f >> 5) & 1, ln = f & 31;
      int row = 4 * j + v + 2 * (ln >> 4);
      int col = ln & 15;
      base[f]       =  ur[col][row];   // B[row][col] = U[col][row]
      base[256 + f] =  ui[col][row];
      base[512 + f] = -ui[col][row];
    }
  }
}

// ---------------- device helpers for the main kernel ----------------
__device__ __forceinline__ void load_b(const float* mb, v2f br[4], v2f bi[4], v2f bn[4], int lane) {
#pragma unroll
  for (int j = 0; j < 4; ++j) {
    br[j][0] = mb[j * 64 + lane];       br[j][1] = mb[j * 64 + 32 + lane];
    bi[j][0] = mb[256 + j * 64 + lane]; bi[j][1] = mb[256 + j * 64 + 32 + lane];
    bn[j][0] = mb[512 + j * 64 + lane]; bn[j][1] = mb[512 + j * 64 + 32 + lane];
  }
}

__device__ __forceinline__ void cplx_gemm(const v2f ar[4], const v2f ai[4],
                                          const v2f br[4], const v2f bi[4], const v2f bn[4],
                                          v8f& dr, v8f& di) {
  dr = v8f{}; di = v8f{};
#pragma unroll
  for (int j = 0; j < 4; ++j) dr = wmma4(ar[j], br[j], dr);
#pragma unroll
  for (int j = 0; j < 4; ++j) dr = wmma4(ai[j], bn[j], dr);  // -Ai*Bi via pre-negated B
#pragma unroll
  for (int j = 0; j < 4; ++j) di = wmma4(ar[j], bi[j], di);
#pragma unroll
  for (int j = 0; j < 4; ++j) di = wmma4(ai[j], br[j], di);
}

// psi' = U_L x psi  on the left nibble; psi viewed as P[16][32], computed as D = P^T_block * U^T
__device__ __forceinline__ void gemm_left(float* pre, float* pim, const float* mb, int lane, int cb) {
  int c = lane & 15, hh = (lane >> 4) << 1;
  v2f ar[4], ai[4], br[4], bi[4], bn[4];
#pragma unroll
  for (int j = 0; j < 4; ++j) {
    int a0 = (4 * j + hh) * 32 + cb * 16 + c;   // A[c][k] = psi[k*32 + cb*16 + c]
    ar[j][0] = pre[a0]; ar[j][1] = pre[a0 + 32];
    ai[j][0] = pim[a0]; ai[j][1] = pim[a0 + 32];
  }
  load_b(mb, br, bi, bn, lane);
  v8f dr, di;
  cplx_gemm(ar, ai, br, bi, bn, dr, di);
  int sb = (lane & 15) * 32 + cb * 16 + 8 * (lane >> 4);  // D[c][m] -> psi[m*32 + cb*16 + c]
#pragma unroll
  for (int v = 0; v < 8; ++v) { pre[sb + v] = dr[v]; pim[sb + v] = di[v]; }
}

// psi' = psi x U_R^T on the right nibble; psi viewed as Q[32][16], D = Q_block * U^T
__device__ __forceinline__ void gemm_right(float* pre, float* pim, const float* mb, int lane, int rb) {
  int r = lane & 15, hh = (lane >> 4) << 1;
  int rowb = (rb * 16 + r) * 16;
  v2f ar[4], ai[4], br[4], bi[4], bn[4];
#pragma unroll
  for (int j = 0; j < 4; ++j) {
    int a0 = rowb + 4 * j + hh;                  // A[r][k] = psi[(rb*16+r)*16 + k]
    ar[j][0] = pre[a0]; ar[j][1] = pre[a0 + 1];
    ai[j][0] = pim[a0]; ai[j][1] = pim[a0 + 1];
  }
  load_b(mb, br, bi, bn, lane);
  v8f dr, di;
  cplx_gemm(ar, ai, br, bi, bn, dr, di);
#pragma unroll
  for (int v = 0; v < 8; ++v) {
    int a = (rb * 16 + v + 8 * (lane >> 4)) * 16 + (lane & 15);  // D[r][n] -> psi[(rb*16+r)*16+n]
    pre[a] = dr[v]; pim[a] = di[v];
  }
}

// 2x2 complex gate g on bit position p (256 disjoint pairs, 8 per lane)
__device__ __forceinline__ void apply2x2(float* pre, float* pim, const float* g, int p, int lane) {
#pragma unroll
  for (int e = 0; e < 8; ++e) {
    int t = e * 32 + lane;
    int k0 = ((t >> p) << (p + 1)) | (t & ((1 << p) - 1));
    int k1 = k0 | (1 << p);
    float are = pre[k0], aim = pim[k0], bre = pre[k1], bim = pim[k1];
    pre[k0] = g[0] * are - g[1] * aim + g[2] * bre - g[3] * bim;
    pim[k0] = g[0] * aim + g[1] * are + g[2] * bim + g[3] * bre;
    pre[k1] = g[4] * are - g[5] * aim + g[6] * bre - g[7] * bim;
    pim[k1] = g[4] * aim + g[5] * are + g[6] * bim + g[7] * bre;
  }
}

// ---------------- main kernel: one wave per statevector ----------------
__global__ __launch_bounds__(256) void qc_main_kernel(const float* __restrict__ x,
                                                      const float* __restrict__ fc_b,
                                                      const float* __restrict__ ws,
                                                      float* __restrict__ out, int B) {
  __shared__ float s_re[WAVES][DIM];
  __shared__ float s_im[WAVES][DIM];
  __shared__ float s_mat[3072];
  __shared__ float s_rot[32];
  int tid = threadIdx.x;
  for (int i = tid; i < 3072; i += 256) s_mat[i] = ws[WS_MAT + i];
  if (tid < 32) s_rot[tid] = ws[WS_ROT + tid];
  __syncthreads();

  int lane = tid & 31, wv = tid >> 5;
  int b = blockIdx.x * WAVES + wv;
  if (b >= B) return;  // whole wave exits together; no barriers after this point
  float* pre = s_re[wv];
  float* pim = s_im[wv];

  // ---- initial product state: psi = (tensor_q RX(x_q)) |0...0> ----
  float cc[NQ], ss[NQ];
#pragma unroll
  for (int q = 0; q < NQ; ++q) {
    float a = 0.5f * x[b * NQ + q];
    cc[q] = cosf(a); ss[q] = sinf(a);
  }
#pragma unroll
  for (int e = 0; e < 16; ++e) {
    int k = e * 32 + lane;
    float r = 1.f; int pop = 0;
#pragma unroll
    for (int q = 0; q < NQ; ++q) {
      int bit = (k >> (8 - q)) & 1;
      r *= bit ? ss[q] : cc[q];
      pop += bit;
    }
    int pm = pop & 3;  // psi[k] = r * (-i)^pop
    pre[k] = (pm == 0) ? r : ((pm == 2) ? -r : 0.f);
    pim[k] = (pm == 1) ? -r : ((pm == 3) ? r : 0.f);
  }

  // ---- two entangling layers ----
  for (int layer = 0; layer < 2; ++layer) {
    const float* mbL = s_mat + (layer * 2 + 0) * 768;
    gemm_left(pre, pim, mbL, lane, 0);
    gemm_left(pre, pim, mbL, lane, 1);
    // CX34 then CX45: pure index permutation, applied as gather new[k] = old[cx34(cx45(k))]
    {
      float tr[16], ti[16];
#pragma unroll
      for (int e = 0; e < 16; ++e) {
        int k = e * 32 + lane;
        int k1 = k ^ (((k >> 4) & 1) << 3);    // cx45: control bit4 -> flip bit3
        int k2 = k1 ^ (((k1 >> 5) & 1) << 4);  // cx34: control bit5 -> flip bit4
        tr[e] = pre[k2]; ti[e] = pim[k2];
      }
#pragma unroll
      for (int e = 0; e < 16; ++e) { int k = e * 32 + lane; pre[k] = tr[e]; pim[k] = ti[e]; }
    }
    apply2x2(pre, pim, s_rot + layer * 16 + 0, 5, lane);  // rot on qubit 3 (bit 5)
    apply2x2(pre, pim, s_rot + layer * 16 + 8, 4, lane);  // rot on qubit 4 (bit 4)
    const float* mbR = s_mat + (layer * 2 + 1) * 768;
    gemm_right(pre, pim, mbR, lane, 0);
    gemm_right(pre, pim, mbR, lane, 1);
  }

  // ---- readout: sigmoid(sum_k |psi_k|^2 * wc[k] + fc_b) ----
  float acc = 0.f;
#pragma unroll
  for (int e = 0; e < 16; ++e) {
    int k = e * 32 + lane;
    float re = pre[k], im = pim[k];
    acc += (re * re + im * im) * ws[k];
  }
#pragma unroll
  for (int off = 16; off > 0; off >>= 1) acc += __shfl_xor(acc, off, 32);
  if (lane == 0) out[b] = 1.f / (1.f + expf(-(acc + fc_b[0])));
}

extern "C" void kernel_launch(void* const* d_in, const int* in_sizes, int n_in,
                              void* d_out, int out_size, void* d_ws, size_t ws_size,
                              hipStream_t stream) {
  const float* x    = (const float*)d_in[0];  // [B,9]
  const float* w    = (const float*)d_in[1];  // [4,9]
  const float* fc_w = (const float*)d_in[2];  // [1,9]
  const float* fc_b = (const float*)d_in[3];  // [1]
  float* ws  = (float*)d_ws;                  // needs 3616 floats (~14.5 KB)
  float* out = (float*)d_out;                 // [B]
  int B = in_sizes[0] / NQ;

  hipLaunchKernelGGL(qc_setup_kernel, dim3(1), dim3(256), 0, stream, w, fc_w, ws);
  int blocks = (B + WAVES - 1) / WAVES;
  hipLaunchKernelGGL(qc_main_kernel, dim3(blocks), dim3(256), 0, stream, x, fc_b, ws, out, B);
}